// Model_38113539785436
// MI455X (gfx1250) — compile-verified
//
#include <hip/hip_runtime.h>

// ChaCha20 single-block keystream on gfx1250 (MI455X).
// One wave32. Inputs fetched via uniform scalar loads; the 16-word working
// state is pinned into VGPRs so the 20-round chain runs on the VALU where
// each 32-bit rotate is a single v_alignbit_b32 (vs 3-op SALU rotate).

__device__ __forceinline__ unsigned int rotl32(unsigned int x, int n) {
    // On VGPR operands the backend selects v_alignbit_b32 (funnel rotate).
    return (x << n) | (x >> (32 - n));
}

#define QR(a, b, c, d)                              \
    a += b; d ^= a; d = rotl32(d, 16);              \
    c += d; b ^= c; b = rotl32(b, 12);              \
    a += b; d ^= a; d = rotl32(d, 8);               \
    c += d; b ^= c; b = rotl32(b, 7);

__global__ __launch_bounds__(32) void chacha20_block_kernel(
    const unsigned int* __restrict__ key,      // 8 words
    const unsigned int* __restrict__ nonce,    // 3 words
    const unsigned int* __restrict__ counter,  // 1 word
    float* __restrict__ out)                   // 16 keystream words as f32
{
    // Initial state (wave-uniform; compiler uses s_load_b256 / s_load_b96).
    const unsigned int s0  = 0x61707865u, s1 = 0x3320646eu,
                       s2  = 0x79622d32u, s3 = 0x6b206574u;
    const unsigned int s4  = key[0],  s5  = key[1],  s6  = key[2],  s7  = key[3];
    const unsigned int s8  = key[4],  s9  = key[5],  s10 = key[6],  s11 = key[7];
    const unsigned int s12 = counter[0];
    const unsigned int s13 = nonce[0], s14 = nonce[1], s15 = nonce[2];

    unsigned int x0 = s0,  x1 = s1,  x2 = s2,  x3 = s3;
    unsigned int x4 = s4,  x5 = s5,  x6 = s6,  x7 = s7;
    unsigned int x8 = s8,  x9 = s9,  x10 = s10, x11 = s11;
    unsigned int x12 = s12, x13 = s13, x14 = s14, x15 = s15;

    // Pin the working state into VGPRs: prevents SALU scalarization and
    // makes every rotate a single v_alignbit_b32. All 32 lanes compute
    // redundantly (same latency as one lane; EXEC stays all-1s).
    asm("" : "+v"(x0), "+v"(x1), "+v"(x2),  "+v"(x3),
             "+v"(x4), "+v"(x5), "+v"(x6),  "+v"(x7),
             "+v"(x8), "+v"(x9), "+v"(x10), "+v"(x11),
             "+v"(x12), "+v"(x13), "+v"(x14), "+v"(x15));

    // 10 double rounds, fully unrolled. Column rounds then diagonal rounds;
    // 4 independent QR chains per half-round give the VALU full ILP.
#pragma unroll
    for (int r = 0; r < 10; ++r) {
        // column rounds
        QR(x0, x4, x8,  x12)
        QR(x1, x5, x9,  x13)
        QR(x2, x6, x10, x14)
        QR(x3, x7, x11, x15)
        // diagonal rounds
        QR(x0, x5, x10, x15)
        QR(x1, x6, x11, x12)
        QR(x2, x7, x8,  x13)
        QR(x3, x4, x9,  x14)
    }

    if (threadIdx.x == 0) {
        // keystream = (x + s) mod 2^32, emitted as f32 (harness output dtype);
        // contiguous stores coalesce into four global_store_b128.
        out[0]  = (float)(x0  + s0);   out[1]  = (float)(x1  + s1);
        out[2]  = (float)(x2  + s2);   out[3]  = (float)(x3  + s3);
        out[4]  = (float)(x4  + s4);   out[5]  = (float)(x5  + s5);
        out[6]  = (float)(x6  + s6);   out[7]  = (float)(x7  + s7);
        out[8]  = (float)(x8  + s8);   out[9]  = (float)(x9  + s9);
        out[10] = (float)(x10 + s10);  out[11] = (float)(x11 + s11);
        out[12] = (float)(x12 + s12);  out[13] = (float)(x13 + s13);
        out[14] = (float)(x14 + s14);  out[15] = (float)(x15 + s15);
    }
}

extern "C" void kernel_launch(void* const* d_in, const int* in_sizes, int n_in,
                              void* d_out, int out_size, void* d_ws, size_t ws_size,
                              hipStream_t stream) {
    (void)in_sizes; (void)n_in; (void)out_size; (void)d_ws; (void)ws_size;
    const unsigned int* key     = (const unsigned int*)d_in[0];
    const unsigned int* nonce   = (const unsigned int*)d_in[1];
    const unsigned int* counter = (const unsigned int*)d_in[2];
    float* out = (float*)d_out;

    // Single wave32: the whole problem is one 16-word block; any more
    // parallelism only adds launch/sync overhead.
    chacha20_block_kernel<<<1, 32, 0, stream>>>(key, nonce, counter, out);
}